// DiffusionLayer_81990925681338
// MI455X (gfx1250) — compile-verified
//
#include <hip/hip_runtime.h>
#include <hip/hip_bf16.h>
#include <math.h>

// Problem constants (match reference)
#define BB   32
#define CC   3
#define HH   256
#define WW   256
#define HWSZ (HH*WW)              // 65536
#define CHW  (CC*HH*WW)           // 196608
#define BCHW (BB*CC*HH*WW)        // 6291456
#define NROW (CC*HH)              // 768 coefficient rows (x dir)
#define DTQ  0.01f
#define EPSQ 1e-06f

typedef __attribute__((ext_vector_type(2))) float v2f;
typedef __attribute__((ext_vector_type(8))) float v8f;

// ---------------------------------------------------------------------------
// Thomas factorization along W for one (c,h) row; shared across batch (32x
// fewer divides). Output stored TRANSPOSED: arr[j*NROW + row] so both these
// writes and the x-solver's reads are coalesced.
// ---------------------------------------------------------------------------
__global__ void precomp_x(const float* __restrict__ base, const float* __restrict__ tc,
                          float t, float dt_eff,
                          float* __restrict__ aT, float* __restrict__ csT,
                          float* __restrict__ invT)
{
    int row = blockIdx.x * blockDim.x + threadIdx.x;   // row = c*HH + h
    if (row >= NROW) return;
    const int o = row * WW;
    float al_m1 = fmaxf(base[o] + tc[o] * t, EPSQ);    // edge replicate
    float al_0  = al_m1;
    float cs_prev = 0.0f;
    for (int j = 0; j < WW; ++j) {
        float al_p1 = (j + 1 < WW) ? fmaxf(base[o+j+1] + tc[o+j+1] * t, EPSQ) : al_0;
        float coeff = (al_m1 + al_0 + al_p1) * (1.0f/3.0f) * dt_eff;
        float aj = -coeff;
        float bj = (j == 0 || j == WW-1) ? (1.0f + coeff) : (1.0f + 2.0f*coeff);
        float denom = (j == 0) ? (bj + EPSQ) : fmaf(-aj, cs_prev, bj) + EPSQ;
        float inv = 1.0f / denom;
        float cs = (-coeff) * inv;
        size_t idx = (size_t)j * NROW + row;           // transposed, coalesced
        aT[idx] = aj; csT[idx] = cs; invT[idx] = inv;
        cs_prev = cs;
        al_m1 = al_0; al_0 = al_p1;
    }
}

// Factorization along H for one (c,w) column (beta swapped (1,2) in the
// reference => smoothing along original H). Stored in (c,h,w) layout: the
// y-solver reads it coalesced as-is.
__global__ void precomp_y(const float* __restrict__ base, const float* __restrict__ tc,
                          float t, float dt_eff,
                          float* __restrict__ a_out, float* __restrict__ cs_out,
                          float* __restrict__ inv_out)
{
    int row = blockIdx.x * blockDim.x + threadIdx.x;   // row = c*WW + w
    if (row >= CC * WW) return;
    const int c = row / WW, w = row % WW;
    const size_t o = (size_t)c * HWSZ + w;             // element j at o + j*WW
    float al_m1 = fmaxf(base[o] + tc[o] * t, EPSQ);
    float al_0  = al_m1;
    float cs_prev = 0.0f;
    for (int j = 0; j < HH; ++j) {
        size_t ip = o + (size_t)(j+1) * WW;
        float al_p1 = (j + 1 < HH) ? fmaxf(base[ip] + tc[ip] * t, EPSQ) : al_0;
        float coeff = (al_m1 + al_0 + al_p1) * (1.0f/3.0f) * dt_eff;
        float aj = -coeff;
        float bj = (j == 0 || j == HH-1) ? (1.0f + coeff) : (1.0f + 2.0f*coeff);
        float denom = (j == 0) ? (bj + EPSQ) : fmaf(-aj, cs_prev, bj) + EPSQ;
        float inv = 1.0f / denom;
        float cs = (-coeff) * inv;
        size_t idx = o + (size_t)j * WW;
        a_out[idx] = aj; cs_out[idx] = cs; inv_out[idx] = inv;
        cs_prev = cs;
        al_m1 = al_0; al_0 = al_p1;
    }
}

// ---------------------------------------------------------------------------
// x-direction solve, LDS-tiled with gfx1250 async LDS<->global copies.
// Each wave owns 32 consecutive rows; rows are streamed through a 32x33 LDS
// tile in 32-column chunks. Tile fill/drain uses ASYNCcnt-tracked
// global_load_async_to_lds_b128 / global_store_async_from_lds_b128
// (512 B, fully coalesced per instruction); each lane then walks its own row
// in LDS (stride 33 -> bank-conflict free). Coefficients come from the
// transposed arrays: coalesced VMEM, no staging needed. Fully in place.
// ---------------------------------------------------------------------------
__global__ __launch_bounds__(256)
void solve_x_tiled(float* __restrict__ u,
                   const float* __restrict__ aT,
                   const float* __restrict__ csT,
                   const float* __restrict__ invT)
{
    __shared__ float ldsAll[8 * 32 * 33];
    const int wv   = threadIdx.x >> 5;
    const int lane = threadIdx.x & 31;
    float* tile = &ldsAll[wv * (32 * 33)];

    const int waveG = blockIdx.x * 8 + wv;
    const int r0 = waveG * 32;                  // first data row of this wave
    const int cr = (r0 % NROW) + lane;          // this lane's coefficient row
    // b128 tile-copy role: 4 rows x 8 lane-quads per instruction
    const int ri  = lane >> 3;                  // row-within-quad 0..3
    const int c4  = (lane & 7) * 4;             // starting column (16B)

    // ---------------- forward sweep ----------------
    float ds = 0.0f;                            // j==0: (d - a*0)*inv = d*inv
    for (int jc = 0; jc < 8; ++jc) {
        const int j0 = jc * 32;
        #pragma unroll
        for (int q = 0; q < 8; ++q) {
            const int i = q * 4 + ri;
            unsigned long long ga =
                (unsigned long long)(uintptr_t)(u + (size_t)(r0 + i) * WW + j0 + c4);
            unsigned la = (unsigned)(uintptr_t)&tile[i * 33 + c4];
            asm volatile("global_load_async_to_lds_b128 %0, %1, off"
                         :: "v"(la), "v"(ga) : "memory");
        }
        asm volatile("s_wait_asynccnt 0x0" ::: "memory");
        #pragma unroll
        for (int jj = 0; jj < 32; ++jj) {
            const int j = j0 + jj;
            float aj  = aT  [(size_t)j * NROW + cr];
            float inv = invT[(size_t)j * NROW + cr];
            float d   = tile[lane * 33 + jj];
            ds = (d - aj * ds) * inv;           // d*[j]
            tile[lane * 33 + jj] = ds;
        }
        asm volatile("s_wait_dscnt 0x0" ::: "memory");   // tile writes done
        #pragma unroll
        for (int q = 0; q < 8; ++q) {
            const int i = q * 4 + ri;
            unsigned long long ga =
                (unsigned long long)(uintptr_t)(u + (size_t)(r0 + i) * WW + j0 + c4);
            unsigned la = (unsigned)(uintptr_t)&tile[i * 33 + c4];
            asm volatile("global_store_async_from_lds_b128 %0, %1, off"
                         :: "v"(ga), "v"(la) : "memory");
        }
        asm volatile("s_wait_asynccnt 0x0" ::: "memory"); // before tile reuse
    }

    // ---------------- backward sweep ----------------
    float x = ds;                               // x[N-1] = d*[N-1]
    for (int jc = 7; jc >= 0; --jc) {
        const int j0 = jc * 32;
        if (jc != 7) {                          // chunk 7 tile still resident
            #pragma unroll
            for (int q = 0; q < 8; ++q) {
                const int i = q * 4 + ri;
                unsigned long long ga =
                    (unsigned long long)(uintptr_t)(u + (size_t)(r0 + i) * WW + j0 + c4);
                unsigned la = (unsigned)(uintptr_t)&tile[i * 33 + c4];
                asm volatile("global_load_async_to_lds_b128 %0, %1, off"
                             :: "v"(la), "v"(ga) : "memory");
            }
            asm volatile("s_wait_asynccnt 0x0" ::: "memory");
        }
        #pragma unroll
        for (int jj = 31; jj >= 0; --jj) {
            const int j = j0 + jj;
            if (j < WW - 1) {                   // last element already final
                float cs    = csT[(size_t)j * NROW + cr];
                float dstar = tile[lane * 33 + jj];
                x = fmaf(-cs, x, dstar);        // x[j] = d*[j] - c*[j]*x[j+1]
                tile[lane * 33 + jj] = x;
            }
        }
        asm volatile("s_wait_dscnt 0x0" ::: "memory");
        #pragma unroll
        for (int q = 0; q < 8; ++q) {
            const int i = q * 4 + ri;
            unsigned long long ga =
                (unsigned long long)(uintptr_t)(u + (size_t)(r0 + i) * WW + j0 + c4);
            unsigned la = (unsigned)(uintptr_t)&tile[i * 33 + c4];
            asm volatile("global_store_async_from_lds_b128 %0, %1, off"
                         :: "v"(ga), "v"(la) : "memory");
        }
        asm volatile("s_wait_asynccnt 0x0" ::: "memory");
    }
}

// ---------------------------------------------------------------------------
// y-direction solve: lane = column w -> every global access is naturally
// coalesced; no staging needed. Fully in place.
// ---------------------------------------------------------------------------
__global__ void solve_y(float* __restrict__ u,
                        const float* __restrict__ a_arr,
                        const float* __restrict__ cs_arr,
                        const float* __restrict__ inv_arr)
{
    int tid = blockIdx.x * blockDim.x + threadIdx.x;   // 0 .. BB*CC*WW-1
    if (tid >= BB * CC * WW) return;
    int b  = tid / (CC * WW);
    int cw = tid % (CC * WW);
    int c = cw / WW, w = cw % WW;
    size_t ro = (size_t)b * CHW + (size_t)c * HWSZ + w;
    size_t co = (size_t)c * HWSZ + w;

    float ds = u[ro] * inv_arr[co];
    u[ro] = ds;
    size_t ru = ro, rc = co;
    for (int j = 1; j < HH; ++j) {
        ru += WW; rc += WW;
        if ((j & 63) == 0 && j + 64 < HH)
            __builtin_prefetch(&u[ru + (size_t)64 * WW], 0, 1);
        ds = (u[ru] - a_arr[rc] * ds) * inv_arr[rc];
        u[ru] = ds;
    }
    float x = ds;
    for (int j = HH - 2; j >= 0; --j) {
        ru -= WW; rc -= WW;
        x = fmaf(-cs_arr[rc], x, u[ru]);
        u[ru] = x;
    }
}

// ---------------------------------------------------------------------------
// Channel coupling out[d] = sum_c M[d,c]*u[c] via V_WMMA_F32_16X16X4_F32.
// Branch-free operand construction (unconditional loads + cndmask selects)
// keeps EXEC all-1s through the WMMA. Last step folds in the sigmoid blend.
// ---------------------------------------------------------------------------
__global__ __launch_bounds__(256)
void couple_wmma(const float* __restrict__ src, float* __restrict__ dst,
                 const float* __restrict__ Mmat, const float* __restrict__ skipw,
                 const float* __restrict__ orig, int do_blend)
{
    const int wave = blockIdx.x * (blockDim.x >> 5) + (threadIdx.x >> 5);
    const int lane = threadIdx.x & 31;
    const int n  = lane & 15;
    const int hi = lane >> 4;
    const int k0 = hi * 2;                       // K pair this half-wave holds
    const int p  = wave * 16 + n;                // flattened (b,hw) pixel
    const int b  = p >> 16;                      // / HWSZ
    const int hw = p & (HWSZ - 1);
    const size_t bo = (size_t)b * CHW;

    // A: rows = out channel (3 of 16), cols = in channel (3 of 4), zero-padded
    const int mrow = (n < 3) ? n : 0;
    const int k1   = (k0 + 1 < 3) ? (k0 + 1) : 0;
    float ax = Mmat[mrow * 3 + k0];
    float ay = Mmat[mrow * 3 + k1];
    v2f A;
    A.x = (n < 3) ? ax : 0.0f;
    A.y = (n < 3 && k0 + 1 < 3) ? ay : 0.0f;
    // B: row K = input channel, col N = pixel (unconditional loads, select)
    float bx = src[bo + (size_t)k0 * HWSZ + hw];
    float by = src[bo + (size_t)k1 * HWSZ + hw];
    v2f Bv;
    Bv.x = bx;
    Bv.y = (k0 + 1 < 3) ? by : 0.0f;

    v8f Cv = {};
    v8f D = __builtin_amdgcn_wmma_f32_16x16x4_f32(
        /*neg_a=*/false, A, /*neg_b=*/false, Bv,
        /*c_mod=*/(short)0, Cv, /*reuse_a=*/false, /*reuse_b=*/false);

    if (lane < 16) {
        size_t i0 = bo + hw;
        if (do_blend) {
            float s = 1.0f / (1.0f + expf(-skipw[0]));
            float t1 = 1.0f - s;
            dst[i0]          = s * orig[i0]          + t1 * D[0];
            dst[i0 + HWSZ]   = s * orig[i0 + HWSZ]   + t1 * D[1];
            dst[i0 + 2*HWSZ] = s * orig[i0 + 2*HWSZ] + t1 * D[2];
        } else {
            dst[i0]          = D[0];
            dst[i0 + HWSZ]   = D[1];
            dst[i0 + 2*HWSZ] = D[2];
        }
    }
}

// ---------------------------------------------------------------------------
extern "C" void kernel_launch(void* const* d_in, const int* in_sizes, int n_in,
                              void* d_out, int out_size, void* d_ws, size_t ws_size,
                              hipStream_t stream)
{
    const float* u_in       = (const float*)d_in[0];
    const float* alpha_base = (const float*)d_in[1];
    const float* beta_base  = (const float*)d_in[2];
    const float* alpha_tc   = (const float*)d_in[3];
    const float* beta_tc    = (const float*)d_in[4];
    const float* Mmat       = (const float*)d_in[5];
    const float* skipw      = (const float*)d_in[6];
    float* out = (float*)d_out;

    float* ws     = (float*)d_ws;
    float* u_work = ws;                       // 25.2 MB working copy (L2-resident)
    float* xa   = u_work + BCHW;              // transposed x factorization
    float* xcs  = xa   + CHW;
    float* xinv = xcs  + CHW;
    float* ya   = xinv + CHW;                 // y factorization, (c,h,w) layout
    float* ycs  = ya   + CHW;
    float* yinv = ycs  + CHW;

    hipMemcpyAsync(u_work, u_in, (size_t)BCHW * sizeof(float),
                   hipMemcpyDeviceToDevice, stream);

    const int xBlocks    = (BB * CC * HH) / (32 * 8);      // 96 (wave = 32 rows)
    const int yBlocks    = (BB * CC * WW) / 256;           // 96
    const int coupBlocks = (BB * HWSZ / 16) / 8;           // 16384

    double t = 0.0;
    for (int step = 0; step < 10; ++step) {
        // x-direction, dt/2, alpha(t)
        precomp_x<<<3, 256, 0, stream>>>(alpha_base, alpha_tc, (float)t, DTQ * 0.5f,
                                         xa, xcs, xinv);
        solve_x_tiled<<<xBlocks, 256, 0, stream>>>(u_work, xa, xcs, xinv);
        t += 0.005;
        // y-direction, dt, beta(t+dt/2)
        precomp_y<<<3, 256, 0, stream>>>(beta_base, beta_tc, (float)t, DTQ,
                                         ya, ycs, yinv);
        solve_y<<<yBlocks, 256, 0, stream>>>(u_work, ya, ycs, yinv);
        t += 0.005;
        // x-direction, dt/2, alpha(t+dt)
        precomp_x<<<3, 256, 0, stream>>>(alpha_base, alpha_tc, (float)t, DTQ * 0.5f,
                                         xa, xcs, xinv);
        solve_x_tiled<<<xBlocks, 256, 0, stream>>>(u_work, xa, xcs, xinv);
        // channel coupling (WMMA); fold skip-blend into the final step
        const int last = (step == 9);
        couple_wmma<<<coupBlocks, 256, 0, stream>>>(u_work, last ? out : u_work,
                                                    Mmat, skipw, u_in, last);
    }
}